// DeferredRender_51780125721348
// MI455X (gfx1250) — compile-verified
//
#include <hip/hip_runtime.h>
#include <math.h>

// ---------------------------------------------------------------------------
// DeferredRender: 4-level multi-resolution bilinear grid_sample, 8 channels.
//   out[c,p] = sum_{tex} sum_{4 corners} w_corner * tex[c, yc, xc]
// Memory-bound gather. Strategy:
//   1) TDM-assisted pre-pass: rewrite each [8,S,S] texture channels-last into
//      d_ws ([S*S, 8]) so the hot gather reads 2x float4 per corner.
//      Tiles are DMA'd into LDS with tensor_load_to_lds (TENSORcnt), then
//      stored out coalesced (32B per thread).
//   2) Gather kernel: 1 thread/pixel, 4 textures x 4 corners, branchless
//      (clamped coords, weight zeroed when invalid) so all 16 corner fetches
//      can issue as load clauses; fmaf accum; 8 coalesced plane stores.
// ---------------------------------------------------------------------------

typedef __attribute__((ext_vector_type(4))) unsigned int v4u;
typedef __attribute__((ext_vector_type(8))) int          v8i;
typedef __attribute__((ext_vector_type(4))) int          v4i;

// Channels-last transpose: src [8, N] (flat planes) -> dst [N, 8].
// One block handles 256 consecutive pixels. Wave 0 issues a TDM descriptor
// that DMAs an 8-row x 256-element tile (row stride N) into LDS, then all
// 256 threads write their pixel's 8 channels contiguously.
__global__ __launch_bounds__(256) void dr_transpose_cl(
    const float* __restrict__ src, float* __restrict__ dst, int N)
{
  __shared__ float lds[8 * 256];
  const int t    = threadIdx.x;
  const int pix0 = blockIdx.x * 256;

#if defined(__gfx1250__) && __has_builtin(__builtin_amdgcn_tensor_load_to_lds)
  if (t < 32) {  // one wave issues the tensor DMA (EXEC ignored by TDM)
    unsigned long long gaddr =
        (unsigned long long)(const void*)src + (unsigned long long)pix0 * 4ull;
    unsigned Nu = (unsigned)N;

    // D# group 0: count=1 | lds_addr=0 | global_addr | type=2 ("image")
    v4u g0;
    g0.x = 1u;                                   // count=1, user mode
    g0.y = 0u;                                   // lds_addr = 0 (tile at LDS base)
    g0.z = (unsigned)(gaddr & 0xFFFFFFFFull);    // global_addr[31:0]
    g0.w = (unsigned)((gaddr >> 32) & 0x1FFFFFFull) | (2u << 30);  // [56:32] | type=2

    // D# group 1: data_size=4B, tensor [dim0=N, dim1=8], tile [256 x 8],
    //             dim0 stride = N elements.
    v8i g1;
    g1.s0 = (int)(2u << 16);                     // data_size=2 (4 bytes)
    g1.s1 = (int)((Nu & 0xFFFFu) << 16);         // tensor_dim0[15:0]
    g1.s2 = (int)((Nu >> 16) | (8u << 16));      // tensor_dim0[31:16] | tensor_dim1=8
    g1.s3 = (int)(256u << 16);                   // tile_dim0 = 256
    g1.s4 = 8;                                   // tile_dim1 = 8, tile_dim2 = 0
    g1.s5 = (int)Nu;                             // tensor_dim0_stride[31:0] = N
    g1.s6 = 0;                                   // stride hi / dim1_stride lo
    g1.s7 = 0;
    v4i gz4 = {0, 0, 0, 0};                      // <=2D tensor: groups 2/3 unused
    v8i gz8 = {0, 0, 0, 0, 0, 0, 0, 0};         // extended group (unused, zero)

    __builtin_amdgcn_tensor_load_to_lds(g0, g1, gz4, gz4, gz8, 0);
    __builtin_amdgcn_s_wait_tensorcnt(0);        // s_wait_tensorcnt 0
  }
  __syncthreads();
#else
  // Fallback staging without TDM (keeps the same LDS layout).
  if (pix0 + t < N) {
    for (int c = 0; c < 8; ++c)
      lds[c * 256 + t] = src[(size_t)c * (size_t)N + (size_t)(pix0 + t)];
  }
  __syncthreads();
#endif

  // The compiler cannot see that tensor_load_to_lds wrote `lds` (the LDS
  // destination is an opaque descriptor field), so without this it folds the
  // LDS reads to undef and deletes the writeback. Escaping the pointer into
  // an opaque asm with a memory clobber forces the loads/stores to exist.
  asm volatile("" : : "r"(&lds[0]) : "memory");

  const int pix = pix0 + t;
  if (pix < N) {
    // Column t of the LDS tile -> contiguous 32B channels-last record.
    float4 lo = make_float4(lds[0 * 256 + t], lds[1 * 256 + t],
                            lds[2 * 256 + t], lds[3 * 256 + t]);
    float4 hi = make_float4(lds[4 * 256 + t], lds[5 * 256 + t],
                            lds[6 * 256 + t], lds[7 * 256 + t]);
    float4* o = (float4*)(dst + (size_t)pix * 8u);
    o[0] = lo;
    o[1] = hi;
  }
}

// Main gather. CL=true: textures channels-last [N,8] (two float4 per corner).
// CL=false: original channels-first [8,S,S] (8 strided loads per corner).
// Branchless corner handling: clamp coordinates (always in-bounds load) and
// zero the corner weight when the un-clamped coordinate was out of range —
// identical semantics to grid_sample(padding_mode='zeros').
template <bool CL>
__global__ __launch_bounds__(256) void dr_gather(
    const float* __restrict__ uv, float* __restrict__ out,
    const float* __restrict__ t0, const float* __restrict__ t1,
    const float* __restrict__ t2, const float* __restrict__ t3,
    int s0, int s1, int s2, int s3, int npix)
{
  const int idx = blockIdx.x * blockDim.x + threadIdx.x;
  if (idx >= npix) return;

  // grid = uv*2-1; align_corners=False => pixel coord = u*S - 0.5
  const float u = uv[idx];
  const float v = uv[(size_t)npix + idx];

  float acc[8] = {0.f, 0.f, 0.f, 0.f, 0.f, 0.f, 0.f, 0.f};
  const float* texs[4]  = {t0, t1, t2, t3};
  const int    sizes[4] = {s0, s1, s2, s3};

#pragma unroll
  for (int ti = 0; ti < 4; ++ti) {
    const float* __restrict__ tex = texs[ti];
    const int s = sizes[ti];
    const float fs = (float)s;

    const float gx  = fmaf(u, fs, -0.5f);
    const float gy  = fmaf(v, fs, -0.5f);
    const float x0f = floorf(gx);
    const float y0f = floorf(gy);
    const int   ix0 = (int)x0f;
    const int   iy0 = (int)y0f;
    const float wx1 = gx - x0f, wy1 = gy - y0f;
    const float wx0 = 1.0f - wx1, wy0 = 1.0f - wy1;

    const float cw[4] = {wx0 * wy0, wx1 * wy0, wx0 * wy1, wx1 * wy1};
    const int   cx[4] = {ix0, ix0 + 1, ix0,     ix0 + 1};
    const int   cy[4] = {iy0, iy0,     iy0 + 1, iy0 + 1};

#pragma unroll
    for (int k = 0; k < 4; ++k) {
      const int  xx    = cx[k], yy = cy[k];
      const bool valid = (xx >= 0) & (xx < s) & (yy >= 0) & (yy < s);
      const float w    = valid ? cw[k] : 0.0f;      // zero padding
      const int  xc    = min(max(xx, 0), s - 1);    // clamped, always in-bounds
      const int  yc    = min(max(yy, 0), s - 1);
      if (CL) {
        const float4* p = (const float4*)(tex + ((size_t)yc * s + xc) * 8u);
        const float4 lo = p[0];
        const float4 hi = p[1];
        acc[0] = fmaf(w, lo.x, acc[0]);
        acc[1] = fmaf(w, lo.y, acc[1]);
        acc[2] = fmaf(w, lo.z, acc[2]);
        acc[3] = fmaf(w, lo.w, acc[3]);
        acc[4] = fmaf(w, hi.x, acc[4]);
        acc[5] = fmaf(w, hi.y, acc[5]);
        acc[6] = fmaf(w, hi.z, acc[6]);
        acc[7] = fmaf(w, hi.w, acc[7]);
      } else {
        const size_t plane = (size_t)s * (size_t)s;
        const size_t base  = (size_t)yc * s + xc;
#pragma unroll
        for (int c = 0; c < 8; ++c)
          acc[c] = fmaf(w, tex[(size_t)c * plane + base], acc[c]);
      }
    }
  }

#pragma unroll
  for (int c = 0; c < 8; ++c)
    out[(size_t)c * (size_t)npix + idx] = acc[c];
}

extern "C" void kernel_launch(void* const* d_in, const int* in_sizes, int n_in,
                              void* d_out, int out_size, void* d_ws, size_t ws_size,
                              hipStream_t stream) {
  const float* uv  = (const float*)d_in[0];
  float*       out = (float*)d_out;

  const float* tex[4];
  int    s[4];
  size_t off[4];
  size_t total_f = 0;
  for (int i = 0; i < 4; ++i) {
    tex[i] = (const float*)d_in[2 + i];
    const long long n = (long long)in_sizes[2 + i] / 8;   // S*S
    s[i]   = (int)(sqrt((double)n) + 0.5);
    off[i] = total_f;
    total_f += (size_t)in_sizes[2 + i];
  }
  const int npix = in_sizes[0] / 2;   // [1,2,H,W] -> H*W pixels
  const int gblk = (npix + 255) / 256;

  float* wsf = (float*)d_ws;
  const bool useCL = (ws_size >= total_f * sizeof(float));

  if (useCL) {
    for (int i = 0; i < 4; ++i) {
      const int N = s[i] * s[i];
      dr_transpose_cl<<<(N + 255) / 256, 256, 0, stream>>>(tex[i], wsf + off[i], N);
    }
    dr_gather<true><<<gblk, 256, 0, stream>>>(
        uv, out, wsf + off[0], wsf + off[1], wsf + off[2], wsf + off[3],
        s[0], s[1], s[2], s[3], npix);
  } else {
    dr_gather<false><<<gblk, 256, 0, stream>>>(
        uv, out, tex[0], tex[1], tex[2], tex[3],
        s[0], s[1], s[2], s[3], npix);
  }
}